// SelectionAttn_62242666054094
// MI455X (gfx1250) — compile-verified
//
#include <hip/hip_runtime.h>
#include <hip/hip_bf16.h>

// Problem constants (from the reference)
#define BS     2
#define NKVH   2
#define GROUP  16
#define NHEAD  32
#define QLEN   2048
#define CLEN   513
#define NOUT   128
#define SELK   16
#define ROWELEMS (GROUP * CLEN)                       // 8208 f32 staged per workgroup
#define HEADSTRIDE 1050624ull                         // QLEN*CLEN elements between heads

typedef unsigned int u32x4 __attribute__((ext_vector_type(4)));
typedef int          i32x4 __attribute__((ext_vector_type(4)));
typedef int          i32x8 __attribute__((ext_vector_type(8)));

#ifndef __has_builtin
#define __has_builtin(x) 0
#endif

// One workgroup per (b, kvh, q) row. 256 threads = 8 wave32.
// Phase 1: ONE Tensor-Data-Mover DMA stages the 16x513 f32 tile (row pitch =
//          head stride) global -> LDS; TENSORcnt tracks completion.
// Phase 2: sum the 16 rows -> score[513] in LDS
// Phase 3: 5-tap / stride-4 average pool -> pooled[128] -> global + LDS
// Phase 4: wave0 iterative top-16 with shfl_xor butterfly reduction
__global__ __launch_bounds__(256) void nsa_select_pool_topk(
    const float* __restrict__ attn, float* __restrict__ out) {
  __shared__ float rows[ROWELEMS];   // 32832 B
  __shared__ float score[CLEN];      //  2052 B
  __shared__ float pooledS[NOUT];    //   512 B

  const int tid = threadIdx.x;
  const int wg  = blockIdx.x;        // == ((b*NKVH + g)*QLEN + q)
  const int q   = wg & (QLEN - 1);
  const int t   = wg >> 11;
  const int g   = t & (NKVH - 1);
  const int b   = t >> 1;

  // Byte address of attn[b, g*GROUP, q, 0]
  const unsigned long long base =
      (((unsigned long long)(b * NHEAD + g * GROUP) * QLEN) + (unsigned)q) * CLEN;
  const unsigned long long gBase = (unsigned long long)(const void*)(attn + base);
  // addrspace(3) -> flat keeps the LDS byte offset in the low 32 bits
  const unsigned ldsBase = (unsigned)(unsigned long long)(void*)rows;

#if __has_builtin(__builtin_amdgcn_tensor_load_to_lds)
  // ---- Phase 1 (TDM): one DMA descriptor moves the whole 16x513 tile ----
  if (tid < 32) {
    // Force descriptor fields uniform (SGPR-resident)
    const unsigned gaLo = (unsigned)__builtin_amdgcn_readfirstlane((int)(gBase & 0xffffffffu));
    const unsigned gaHi = (unsigned)__builtin_amdgcn_readfirstlane((int)(gBase >> 32));
    const unsigned lBase = (unsigned)__builtin_amdgcn_readfirstlane((int)ldsBase);

    // D# group 0: [1:0]=count=1, [63:32]=lds_addr, [120:64]=global_addr, [127:126]=type=2
    u32x4 g0;
    g0[0] = 1u;
    g0[1] = lBase;
    g0[2] = gaLo;
    g0[3] = (gaHi & 0x01ffffffu) | (2u << 30);

    // D# group 1: [17:16]=data_size(2=4B), [79:48]=tensor_dim0=513,
    // [111:80]=tensor_dim1=16, [127:112]=tile_dim0=513, [143:128]=tile_dim1=16,
    // [207:160]=tensor_dim0_stride=1050624 (elements), rest 0
    i32x8 g1 = {};
    g1[0] = (int)(2u << 16);
    g1[1] = (int)(513u << 16);
    g1[2] = (int)(16u << 16);
    g1[3] = (int)(513u << 16);
    g1[4] = 16;
    g1[5] = (int)(unsigned)HEADSTRIDE;

    i32x4 g2 = {};   // 2D tile: groups 2/3 unused
    i32x4 g3 = {};
#if defined(__clang_major__) && (__clang_major__ >= 23)
    i32x8 g4 = {};
    __builtin_amdgcn_tensor_load_to_lds(g0, g1, g2, g3, g4, 0);
#else
    __builtin_amdgcn_tensor_load_to_lds(g0, g1, g2, g3, 0);
#endif
    __builtin_amdgcn_s_wait_tensorcnt(0);
  }
  __syncthreads();
#else
  // ---- Phase 1 (fallback): per-lane async global -> LDS copies ----
  for (int i = tid; i < ROWELEMS; i += 256) {
    const int h = i / CLEN;
    const int c = i - h * CLEN;
    unsigned long long ga =
        gBase + 4ull * ((unsigned long long)h * HEADSTRIDE + (unsigned)c);
    unsigned la = ldsBase + 4u * (unsigned)i;
    asm volatile("global_load_async_to_lds_b32 %0, %1, off"
                 :: "v"(la), "v"(ga)
                 : "memory");
  }
  asm volatile("s_wait_asynccnt 0" ::: "memory");
  __syncthreads();
#endif

  // ---- Phase 2: GQA head-group sum -> score[c] ----
  for (int c = tid; c < CLEN; c += 256) {
    float s = 0.0f;
#pragma unroll
    for (int h = 0; h < GROUP; ++h) s += rows[h * CLEN + c];
    score[c] = s;
  }
  __syncthreads();

  // ---- Phase 3: AvgPool1d(kernel=5, stride=4) -> pooled[o] ----
  if (tid < NOUT) {
    const int o = tid;
    const float* sp = &score[4 * o];
    const float p = (sp[0] + sp[1] + sp[2] + sp[3] + sp[4]) * 0.2f;
    pooledS[o] = p;
    out[(unsigned long long)wg * NOUT + o] = p;
  }
  __syncthreads();

  // ---- Phase 4: top-16 of 128 (wave0 only), stable tie-break on lower index ----
  if (tid < 32) {
    float v[4];
    int   id[4];
#pragma unroll
    for (int j = 0; j < 4; ++j) {
      const int o = tid + j * 32;
      v[j]  = pooledS[o];
      id[j] = o;
    }
    const unsigned long long idxBase =
        (unsigned long long)BS * NKVH * QLEN * NOUT + (unsigned long long)wg * SELK;
    for (int it = 0; it < SELK; ++it) {
      float bv = v[0];
      int   bi = id[0];
#pragma unroll
      for (int j = 1; j < 4; ++j)
        if (v[j] > bv || (v[j] == bv && id[j] < bi)) { bv = v[j]; bi = id[j]; }
      // butterfly: every lane converges to the global (max value, min index)
#pragma unroll
      for (int off = 16; off > 0; off >>= 1) {
        const float ov = __shfl_xor(bv, off, 32);
        const int   oi = __shfl_xor(bi, off, 32);
        if (ov > bv || (ov == bv && oi < bi)) { bv = ov; bi = oi; }
      }
      if (tid == 0) out[idxBase + it] = (float)bi;
#pragma unroll
      for (int j = 0; j < 4; ++j)
        if (id[j] == bi) v[j] = -__builtin_inff();
    }
  }
}

extern "C" void kernel_launch(void* const* d_in, const int* in_sizes, int n_in,
                              void* d_out, int out_size, void* d_ws, size_t ws_size,
                              hipStream_t stream) {
  (void)in_sizes; (void)n_in; (void)d_ws; (void)ws_size; (void)out_size;
  const float* attn = (const float*)d_in[0];   // (2,32,2048,513) f32; q/k/v unused
  float* out = (float*)d_out;                  // pooled (2,2,2048,128) ++ indices (2,2,2048,16)
  dim3 grid(BS * NKVH * QLEN);                 // 8192 workgroups
  nsa_select_pool_topk<<<grid, 256, 0, stream>>>(attn, out);
}